// ASAPBlock_73117523247416
// MI455X (gfx1250) — compile-verified
//
#include <hip/hip_runtime.h>
#include <hip/hip_bf16.h>

// ASAP pooling block for gfx1250 (MI455X).
// fp32 throughout; GEMMs via V_WMMA_F32_16X16X4_F32, 8 waves/block, 16x64 strip
// per wave, B staged in LDS (bank-conflict-free padded stride), K split across
// gridDim.z with f32 atomic accumulation for the deep S^T*(Adj*S) GEMM.

#define FDIM 64
#define KSEL 256

#define BM 128        // rows per block (8 waves x 16)
#define BN 64         // cols per block (4 WMMA n-tiles per wave)
#define BK 64         // K chunk staged in LDS
#define LDS_STRIDE 72 // 64+8: rows k and k+2 differ by 144 dwords = 16 mod 64 banks

typedef float v2f __attribute__((ext_vector_type(2)));
typedef float v8f __attribute__((ext_vector_type(8)));

__device__ __forceinline__ void edge_rc(const int* __restrict__ ei, int E, int e, int& r, int& c) {
    if (e < E) { r = ei[e]; c = ei[E + e]; }
    else       { r = e - E; c = e - E; }            // appended self-loops
}

// order-preserving float<->uint encoding for atomicMax over signed floats
__device__ __forceinline__ unsigned enc_f32(float f) {
    unsigned u = __float_as_uint(f);
    return (u & 0x80000000u) ? ~u : (u | 0x80000000u);
}
__device__ __forceinline__ float dec_f32(unsigned k) {
    return (k & 0x80000000u) ? __uint_as_float(k ^ 0x80000000u) : __uint_as_float(~k);
}

__global__ void fill_f32(float* p, float v, int n) {
    int i = blockIdx.x * blockDim.x + threadIdx.x;
    if (i < n) p[i] = v;
}
__global__ void fill_u32(unsigned* p, unsigned v, int n) {
    int i = blockIdx.x * blockDim.x + threadIdx.x;
    if (i < n) p[i] = v;
}

// ---------------- degree / norm ----------------
__global__ void k_deg(const int* __restrict__ ei, int E, float* __restrict__ deg) {
    int e = blockIdx.x * blockDim.x + threadIdx.x;
    if (e < E) atomicAdd(&deg[ei[E + e]], 1.0f);    // col side; self-loops folded into init=1
}
__global__ void k_rsqrt(float* __restrict__ deg, int n) {
    int i = blockIdx.x * blockDim.x + threadIdx.x;
    if (i < n) deg[i] = rsqrtf(deg[i]);
}

// ---------------- fp32 WMMA GEMM ----------------
// C[M,N] = A @ B (+bias) or atomic-accumulate partial products (K split on grid.z).
// A addressed as A[m*rsA + k*csA] (rsA=1,csA=ld gives A^T). B row-major KxN.
// Block: 256 threads = 8 waves; wave w owns rows m0+16w..m0+16w+15, cols n0..n0+63.
// Per k-step: one A fragment feeds 4 WMMAs; B fragments come from LDS.
__global__ __launch_bounds__(256)
void gemm_wmma_f32(const float* __restrict__ A, long rsA, long csA,
                   const float* __restrict__ B, long ldb,
                   float* __restrict__ C, long ldc,
                   int M, int N, int Kdim, int kPerBlock,
                   const float* __restrict__ bias, int accumulate)
{
    __shared__ float Bs[BK * LDS_STRIDE];

    const int tid  = threadIdx.x;
    const int wave = tid >> 5;
    const int lane = tid & 31;
    const int half = lane >> 4;            // 0: K=k,k+1 ; 1: K=k+2,k+3
    const int mn   = lane & 15;

    const long m0 = (long)blockIdx.x * BM + wave * 16;
    const long n0 = (long)blockIdx.y * BN;
    const bool mvalid = (m0 + 16 <= (long)M);          // wave-uniform -> EXEC all-ones

    const int kBeg = blockIdx.z * kPerBlock;
    const int kEnd = min(kBeg + kPerBlock, Kdim);

    v8f acc0 = {}, acc1 = {}, acc2 = {}, acc3 = {};

    for (int k0 = kBeg; k0 < kEnd; k0 += BK) {
        const int kc = min(BK, kEnd - k0);             // multiple of 4 by construction
        // stage B chunk [k0..k0+kc) x [n0..n0+BN) into LDS (coalesced)
        for (int idx = tid; idx < kc * BN; idx += 256) {
            int kr = idx >> 6, nc = idx & 63;
            Bs[kr * LDS_STRIDE + nc] = B[(long)(k0 + kr) * ldb + n0 + nc];
        }
        if (k0 + BK < kEnd && tid < BK)                // prefetch next chunk's rows
            __builtin_prefetch(&B[(long)(k0 + BK + tid) * ldb + n0], 0, 1);
        __syncthreads();

        if (mvalid) {
            const long arow = (m0 + mn) * rsA;
            for (int kk = 0; kk < kc; kk += 4) {
                const int ka = kk + 2 * half;
                v2f a, b0, b1, b2, b3;
                a.x = A[arow + (long)(k0 + ka) * csA];
                a.y = A[arow + (long)(k0 + ka + 1) * csA];
                const float* r0 = &Bs[ka * LDS_STRIDE + mn];
                const float* r1 = &Bs[(ka + 1) * LDS_STRIDE + mn];
                b0.x = r0[0];  b0.y = r1[0];
                b1.x = r0[16]; b1.y = r1[16];
                b2.x = r0[32]; b2.y = r1[32];
                b3.x = r0[48]; b3.y = r1[48];
                acc0 = __builtin_amdgcn_wmma_f32_16x16x4_f32(false, a, false, b0, (short)0, acc0, false, false);
                acc1 = __builtin_amdgcn_wmma_f32_16x16x4_f32(false, a, false, b1, (short)0, acc1, false, false);
                acc2 = __builtin_amdgcn_wmma_f32_16x16x4_f32(false, a, false, b2, (short)0, acc2, false, false);
                acc3 = __builtin_amdgcn_wmma_f32_16x16x4_f32(false, a, false, b3, (short)0, acc3, false, false);
            }
        }
        __syncthreads();
    }

    if (!mvalid) return;
    union { v8f v; float f[8]; } u0, u1, u2, u3;
    u0.v = acc0; u1.v = acc1; u2.v = acc2; u3.v = acc3;
#pragma unroll
    for (int r = 0; r < 8; ++r) {
        long mm = m0 + r + 8 * half;                   // C VGPR r: M=r / r+8 per lane half
        float* crow = C + mm * ldc + n0 + mn;
        float v0 = u0.f[r], v1 = u1.f[r], v2 = u2.f[r], v3 = u3.f[r];
        if (accumulate) {
            atomicAdd(crow +  0, v0);
            atomicAdd(crow + 16, v1);
            atomicAdd(crow + 32, v2);
            atomicAdd(crow + 48, v3);
        } else {
            if (bias) {
                v0 += bias[n0 + mn];      v1 += bias[n0 + 16 + mn];
                v2 += bias[n0 + 32 + mn]; v3 += bias[n0 + 48 + mn];
            }
            crow[0] = v0; crow[16] = v1; crow[32] = v2; crow[48] = v3;
        }
    }
}

// ---------------- GCN aggregation: out[c,f] += dis[r]*dis[c]*h[r,f] ----------------
__global__ void k_gcn_agg(const float* __restrict__ h, const float* __restrict__ dis,
                          const int* __restrict__ ei, int E, int EN,
                          float* __restrict__ out)
{
    long t = (long)blockIdx.x * blockDim.x + threadIdx.x;
    int e = (int)(t >> 6), f = (int)(t & 63);
    if (e >= EN) return;
    int r, c; edge_rc(ei, E, e, r, c);
    float nrm = dis[r] * dis[c];
    atomicAdd(&out[(long)c * FDIM + f], nrm * h[(long)r * FDIM + f]);
}
__global__ void k_bias_relu(float* __restrict__ x, const float* __restrict__ b, int nf) {
    int i = blockIdx.x * blockDim.x + threadIdx.x;
    if (i < nf) x[i] = fmaxf(x[i] + b[i & 63], 0.0f);
}

// ---------------- segment max of x2[row] over col (values >= 0 post-ReLU) ----------------
__global__ void k_segmax(const float* __restrict__ x2, const int* __restrict__ ei,
                         int E, int EN, unsigned* __restrict__ xq_u)
{
    long t = (long)blockIdx.x * blockDim.x + threadIdx.x;
    int e = (int)(t >> 6), f = (int)(t & 63);
    if (e >= EN) return;
    int r, c; edge_rc(ei, E, e, r, c);
    atomicMax(&xq_u[(long)c * FDIM + f], __float_as_uint(x2[(long)r * FDIM + f]));
}

// ---------------- attention score per edge (1 wave / edge) ----------------
__global__ void k_score(const float* __restrict__ xql, const float* __restrict__ x2,
                        const int* __restrict__ ei, int E, int EN,
                        const float* __restrict__ attW, const float* __restrict__ attB,
                        float* __restrict__ score, unsigned* __restrict__ smax_u)
{
    int e = blockIdx.x * 8 + (threadIdx.x >> 5);
    int t = threadIdx.x & 31;
    if (e >= EN) return;
    int r, c; edge_rc(ei, E, e, r, c);
    float p = attW[t]       * xql[(long)c * FDIM + t]
            + attW[t + 32]  * xql[(long)c * FDIM + t + 32]
            + attW[64 + t]  * x2[(long)r * FDIM + t]
            + attW[96 + t]  * x2[(long)r * FDIM + t + 32];
    for (int off = 16; off > 0; off >>= 1) p += __shfl_xor(p, off, 32);
    if (t == 0) {
        float s = p + attB[0];
        s = (s > 0.0f) ? s : 0.2f * s;               // leaky_relu(0.2)
        score[e] = s;
        atomicMax(&smax_u[c], enc_f32(s));
    }
}

__global__ void k_expsum(float* __restrict__ score, const unsigned* __restrict__ smax_u,
                         const int* __restrict__ ei, int E, int EN, float* __restrict__ ssum)
{
    int e = blockIdx.x * blockDim.x + threadIdx.x;
    if (e >= EN) return;
    int r, c; edge_rc(ei, E, e, r, c);
    float ev = __expf(score[e] - dec_f32(smax_u[c]));
    score[e] = ev;
    atomicAdd(&ssum[c], ev);
}

// normalized score -> score2; xc[c,f] += x2[r,f]*s
__global__ void k_soft_xc(const float* __restrict__ score, const float* __restrict__ ssum,
                          const float* __restrict__ x2, const int* __restrict__ ei,
                          int E, int EN, float* __restrict__ score2, float* __restrict__ xc)
{
    long t = (long)blockIdx.x * blockDim.x + threadIdx.x;
    int e = (int)(t >> 6), f = (int)(t & 63);
    if (e >= EN) return;
    int r, c; edge_rc(ei, E, e, r, c);
    float s = score[e] / ssum[c];
    if (f == 0) score2[e] = s;
    atomicAdd(&xc[(long)c * FDIM + f], x2[(long)r * FDIM + f] * s);
}

// per-node scalars: a = xc.le1+le1_b, b = xc.le2, c3 = xc.le3  (1 wave / node)
__global__ void k_fitpre(const float* __restrict__ xc,
                         const float* __restrict__ w1, const float* __restrict__ b1,
                         const float* __restrict__ w2, const float* __restrict__ w3,
                         int n, float* __restrict__ av, float* __restrict__ bv,
                         float* __restrict__ c3)
{
    int v = blockIdx.x * 8 + (threadIdx.x >> 5);
    int t = threadIdx.x & 31;
    if (v >= n) return;
    float x0 = xc[(long)v * FDIM + t], x1 = xc[(long)v * FDIM + t + 32];
    float pa = x0 * w1[t] + x1 * w1[t + 32];
    float pb = x0 * w2[t] + x1 * w2[t + 32];
    float pc = x0 * w3[t] + x1 * w3[t + 32];
    for (int off = 16; off > 0; off >>= 1) {
        pa += __shfl_xor(pa, off, 32);
        pb += __shfl_xor(pb, off, 32);
        pc += __shfl_xor(pc, off, 32);
    }
    if (t == 0) { av[v] = pa + b1[0]; bv[v] = pb; c3[v] = pc; }
}

__global__ void k_aggk(const int* __restrict__ ei, int E, int EN,
                       const float* __restrict__ av, const float* __restrict__ bv,
                       float* __restrict__ agg)
{
    int e = blockIdx.x * blockDim.x + threadIdx.x;
    if (e >= EN) return;
    int r, c; edge_rc(ei, E, e, r, c);
    atomicAdd(&agg[c], av[r] - bv[c]);
}

__global__ void k_fit(const float* __restrict__ agg, const float* __restrict__ c3,
                      const float* __restrict__ b3, int n,
                      float* __restrict__ fit, float* __restrict__ fitc)
{
    int v = blockIdx.x * blockDim.x + threadIdx.x;
    if (v >= n) return;
    float s = agg[v] + c3[v] + b3[0];
    float f = 1.0f / (1.0f + __expf(-s));
    fit[v] = f; fitc[v] = f;
}

// deterministic top-K (stable: ties -> lowest index), single workgroup
__global__ __launch_bounds__(1024)
void k_topk(float* __restrict__ fitc, int n, int* __restrict__ perm, int* __restrict__ inv)
{
    __shared__ float sv[1024];
    __shared__ int   si[1024];
    int t = threadIdx.x;
    for (int v = t; v < n; v += 1024) inv[v] = KSEL;
    __syncthreads();
    for (int k = 0; k < KSEL; ++k) {
        float bvv = -1e30f; int bi = 0x7fffffff;
        for (int v = t; v < n; v += 1024) {
            float f = fitc[v];
            if (f > bvv || (f == bvv && v < bi)) { bvv = f; bi = v; }
        }
        sv[t] = bvv; si[t] = bi;
        __syncthreads();
        for (int off = 512; off > 0; off >>= 1) {
            if (t < off) {
                if (sv[t + off] > sv[t] || (sv[t + off] == sv[t] && si[t + off] < si[t])) {
                    sv[t] = sv[t + off]; si[t] = si[t + off];
                }
            }
            __syncthreads();
        }
        if (t == 0) { int p = si[0]; perm[k] = p; inv[p] = k; fitc[p] = -1e30f; }
        __threadfence_block();
        __syncthreads();
    }
}

__global__ void k_xnew(const float* __restrict__ xc, const float* __restrict__ fit,
                       const int* __restrict__ perm, float* __restrict__ out)
{
    int i = blockIdx.x * blockDim.x + threadIdx.x;
    if (i >= KSEL * FDIM) return;
    int k = i >> 6, f = i & 63;
    int p = perm[k];
    out[i] = xc[(long)p * FDIM + f] * fit[p];
}

// S[r, inv[c]] += score2[e]   (row stride 257; column 256 is the trash bucket)
__global__ void k_S(const int* __restrict__ ei, int E, int EN, const int* __restrict__ inv,
                    const float* __restrict__ score2, float* __restrict__ S)
{
    int e = blockIdx.x * blockDim.x + threadIdx.x;
    if (e >= EN) return;
    int r, c; edge_rc(ei, E, e, r, c);
    atomicAdd(&S[(long)r * (KSEL + 1) + inv[c]], score2[e]);
}

// AS[r, j] += S[c, j]   (one block per edge, 256 lanes = j)
__global__ void k_AS(const int* __restrict__ ei, int E, int EN,
                     const float* __restrict__ S, float* __restrict__ AS)
{
    int e = blockIdx.x;
    if (e >= EN) return;
    int j = threadIdx.x;
    int r, c; edge_rc(ei, E, e, r, c);
    atomicAdd(&AS[(long)r * KSEL + j], S[(long)c * (KSEL + 1) + j]);
}

__global__ void k_diag(float* __restrict__ Anew) {
    int i = blockIdx.x * blockDim.x + threadIdx.x;
    if (i < KSEL) Anew[(long)i * KSEL + i] = 1.0f;
}

static inline int cdiv(long a, long b) { return (int)((a + b - 1) / b); }

extern "C" void kernel_launch(void* const* d_in, const int* in_sizes, int n_in,
                              void* d_out, int out_size, void* d_ws, size_t ws_size,
                              hipStream_t stream)
{
    const float* x      = (const float*)d_in[0];
    const int*   ei     = (const int*)d_in[1];
    const float* W0     = (const float*)d_in[2];
    const float* b0     = (const float*)d_in[3];
    const float* W1     = (const float*)d_in[4];
    const float* b1     = (const float*)d_in[5];
    const float* linW   = (const float*)d_in[6];
    const float* linB   = (const float*)d_in[7];
    const float* attW   = (const float*)d_in[8];
    const float* attB   = (const float*)d_in[9];
    const float* le1W   = (const float*)d_in[10];
    const float* le1B   = (const float*)d_in[11];
    const float* le2W   = (const float*)d_in[12];
    const float* le3W   = (const float*)d_in[13];
    const float* le3B   = (const float*)d_in[14];

    const int N  = in_sizes[0] / FDIM;      // 20000
    const int E  = in_sizes[1] / 2;         // 320000
    const int EN = E + N;                   // with self-loops
    const long NF = (long)N * FDIM;

    // ---- workspace carve-up (fp32 elements) ----
    float* w = (float*)d_ws;
    float*    deg    = w;            w += N;            // becomes dis after rsqrt
    float*    h      = w;            w += NF;           // GEMM temp (reused for xq@linW)
    float*    x1     = w;            w += NF;
    float*    x2     = w;            w += NF;
    unsigned* xq_u   = (unsigned*)w; w += NF;           // bits of nonneg floats
    float*    score  = w;            w += EN;
    float*    score2 = w;            w += EN;
    unsigned* smax_u = (unsigned*)w; w += N;
    float*    ssum   = w;            w += N;
    float*    xc     = w;            w += NF;
    float*    av     = w;            w += N;
    float*    bv     = w;            w += N;
    float*    c3     = w;            w += N;
    float*    agg    = w;            w += N;
    float*    fit    = w;            w += N;
    float*    fitc   = w;            w += N;
    int*      inv    = (int*)w;      w += N;
    float*    S      = w;            w += (long)N * (KSEL + 1);
    float*    AS     = w;            w += (long)N * KSEL;

    float* out_x = (float*)d_out;                        // K*F
    float* out_A = out_x + KSEL * FDIM;                  // K*K
    int*   out_p = (int*)(out_A + KSEL * KSEL);          // K ints (bit-cast)

    const int B = 256;

    // ---- init ----
    fill_f32<<<cdiv(N, B), B, 0, stream>>>(deg, 1.0f, N);       // self-loop degree
    fill_f32<<<cdiv(NF, B), B, 0, stream>>>(x1, 0.0f, (int)NF);
    fill_f32<<<cdiv(NF, B), B, 0, stream>>>(x2, 0.0f, (int)NF);
    fill_u32<<<cdiv(NF, B), B, 0, stream>>>(xq_u, 0u, (int)NF);
    fill_u32<<<cdiv(N, B), B, 0, stream>>>(smax_u, 0x007FFFFFu, N);  // enc(-inf)
    fill_f32<<<cdiv(N, B), B, 0, stream>>>(ssum, 0.0f, N);
    fill_f32<<<cdiv(NF, B), B, 0, stream>>>(xc, 0.0f, (int)NF);
    fill_f32<<<cdiv(N, B), B, 0, stream>>>(agg, 0.0f, N);
    fill_f32<<<cdiv((long)N * (KSEL + 1), B), B, 0, stream>>>(S, 0.0f, N * (KSEL + 1));
    fill_f32<<<cdiv((long)N * KSEL, B), B, 0, stream>>>(AS, 0.0f, N * KSEL);
    fill_f32<<<cdiv(KSEL * KSEL, B), B, 0, stream>>>(out_A, 0.0f, KSEL * KSEL);

    // ---- degree / symmetric norm ----
    k_deg<<<cdiv(E, B), B, 0, stream>>>(ei, E, deg);
    k_rsqrt<<<cdiv(N, B), B, 0, stream>>>(deg, N);               // deg := dis

    dim3 gemmF(cdiv(N, BM), 1, 1);   // feature GEMMs: M=N_nodes, N=K=64

    // ---- GCN layer 1 ----
    gemm_wmma_f32<<<gemmF, 256, 0, stream>>>(x, FDIM, 1, W0, FDIM, h, FDIM,
                                             N, FDIM, FDIM, FDIM, nullptr, 0);
    k_gcn_agg<<<cdiv((long)EN * FDIM, B), B, 0, stream>>>(h, deg, ei, E, EN, x1);
    k_bias_relu<<<cdiv(NF, B), B, 0, stream>>>(x1, b0, (int)NF);

    // ---- GCN layer 2 ----
    gemm_wmma_f32<<<gemmF, 256, 0, stream>>>(x1, FDIM, 1, W1, FDIM, h, FDIM,
                                             N, FDIM, FDIM, FDIM, nullptr, 0);
    k_gcn_agg<<<cdiv((long)EN * FDIM, B), B, 0, stream>>>(h, deg, ei, E, EN, x2);
    k_bias_relu<<<cdiv(NF, B), B, 0, stream>>>(x2, b1, (int)NF);

    // ---- pooling attention ----
    k_segmax<<<cdiv((long)EN * FDIM, B), B, 0, stream>>>(x2, ei, E, EN, xq_u);
    gemm_wmma_f32<<<gemmF, 256, 0, stream>>>((const float*)xq_u, FDIM, 1, linW, FDIM,
                                             h, FDIM, N, FDIM, FDIM, FDIM, linB, 0);
    k_score<<<cdiv(EN, 8), 256, 0, stream>>>(h, x2, ei, E, EN, attW, attB, score, smax_u);
    k_expsum<<<cdiv(EN, B), B, 0, stream>>>(score, smax_u, ei, E, EN, ssum);
    k_soft_xc<<<cdiv((long)EN * FDIM, B), B, 0, stream>>>(score, ssum, x2, ei, E, EN, score2, xc);

    // ---- fitness ----
    k_fitpre<<<cdiv(N, 8), 256, 0, stream>>>(xc, le1W, le1B, le2W, le3W, N, av, bv, c3);
    k_aggk<<<cdiv(EN, B), B, 0, stream>>>(ei, E, EN, av, bv, agg);
    k_fit<<<cdiv(N, B), B, 0, stream>>>(agg, c3, le3B, N, fit, fitc);

    // ---- top-K, outputs ----
    k_topk<<<1, 1024, 0, stream>>>(fitc, N, out_p, inv);
    k_xnew<<<cdiv(KSEL * FDIM, B), B, 0, stream>>>(xc, fit, out_p, out_x);

    // ---- coarsened adjacency: A_new = S^T @ (Adj @ S) ----
    k_S<<<cdiv(EN, B), B, 0, stream>>>(ei, E, EN, inv, score2, S);
    k_AS<<<EN, KSEL, 0, stream>>>(ei, E, EN, S, AS);

    // K-split GEMM: grid.z blocks each own kPerBlock (multiple of 4) of the
    // N_nodes-deep reduction, accumulating via global f32 atomics.
    int kPerBlock = ((cdiv(N, 25) + 3) / 4) * 4;       // 800 for N=20000
    int gz = cdiv(N, kPerBlock);                        // 25
    gemm_wmma_f32<<<dim3(cdiv(KSEL, BM), KSEL / BN, gz), 256, 0, stream>>>(
        S, 1, KSEL + 1,            // A = S^T : A[i,v] = S[v*257 + i]
        AS, KSEL,
        out_A, KSEL,
        KSEL, KSEL, N, kPerBlock, nullptr, 1);
    k_diag<<<1, KSEL, 0, stream>>>(out_A);
}